// StructuredEnergy_12558484373683
// MI455X (gfx1250) — compile-verified
//
#include <hip/hip_runtime.h>
#include <math.h>

#define Bc   64
#define Lc_  512
#define SZ   128
#define HEADSc 8
#define AHc  16
#define NBRc 15
#define KERc 16
#define DEPTHc 3
#define Nn   (Bc * Lc_)      // 32768
#define KVK  157
#define KVKP 160
#define MAXDc 20.0f

typedef float f32x2 __attribute__((ext_vector_type(2)));
typedef float f32x8 __attribute__((ext_vector_type(8)));

__device__ __forceinline__ f32x8 wmma4(f32x2 a, f32x2 b, f32x8 c) {
    // D = A(16x4,f32) * B(4x16,f32) + C(16x16,f32)  -> v_wmma_f32_16x16x4_f32
    return __builtin_amdgcn_wmma_f32_16x16x4_f32(false, a, false, b, (short)0, c, false, false);
}

// ---------------------------------------------------------------- frames ----
__global__ void frames_kernel(const float* __restrict__ tert,
                              float* __restrict__ pos, float* __restrict__ R) {
    int n = blockIdx.x * blockDim.x + threadIdx.x;
    if (n >= Nn) return;
    int nm = n > 0 ? n - 1 : 0;
    int np = n < Nn - 1 ? n + 1 : Nn - 1;
    float pm[3], p0[3], pp[3];
    for (int c = 0; c < 3; ++c) {
        pm[c] = tert[nm * 9 + 3 + c];
        p0[c] = tert[n  * 9 + 3 + c];
        pp[c] = tert[np * 9 + 3 + c];
    }
    pos[n * 3 + 0] = p0[0]; pos[n * 3 + 1] = p0[1]; pos[n * 3 + 2] = p0[2];
    float up[3], un[3];
    for (int c = 0; c < 3; ++c) {
        up[c] = (n == 0)      ? (pp[c] - p0[c]) : (p0[c] - pm[c]);
        un[c] = (n == Nn - 1) ? (p0[c] - pm[c]) : (pp[c] - p0[c]);
    }
    float lu = sqrtf(up[0]*up[0] + up[1]*up[1] + up[2]*up[2]) + 1e-8f;
    float ln = sqrtf(un[0]*un[0] + un[1]*un[1] + un[2]*un[2]) + 1e-8f;
    for (int c = 0; c < 3; ++c) { up[c] /= lu; un[c] /= ln; }
    float b[3], nv[3];
    for (int c = 0; c < 3; ++c) b[c] = up[c] - un[c];
    float lb = sqrtf(b[0]*b[0] + b[1]*b[1] + b[2]*b[2]) + 1e-8f;
    for (int c = 0; c < 3; ++c) b[c] /= lb;
    nv[0] = up[1]*un[2] - up[2]*un[1];
    nv[1] = up[2]*un[0] - up[0]*un[2];
    nv[2] = up[0]*un[1] - up[1]*un[0];
    float lnv = sqrtf(nv[0]*nv[0] + nv[1]*nv[1] + nv[2]*nv[2]) + 1e-8f;
    for (int c = 0; c < 3; ++c) nv[c] /= lnv;
    float c3[3];
    c3[0] = b[1]*nv[2] - b[2]*nv[1];
    c3[1] = b[2]*nv[0] - b[0]*nv[2];
    c3[2] = b[0]*nv[1] - b[1]*nv[0];
    for (int r = 0; r < 3; ++r) {
        R[n * 9 + r * 3 + 0] = b[r];
        R[n * 9 + r * 3 + 1] = nv[r];
        R[n * 9 + r * 3 + 2] = c3[r];
    }
}

// ----------------------------------------------------------------- top-k ----
__global__ void topk_kernel(const float* __restrict__ pos,
                            const float* __restrict__ noise,
                            int* __restrict__ nbr) {
    int i = blockIdx.x;   // node within batch
    int b = blockIdx.y;   // batch
    __shared__ float cl[Lc_];
    __shared__ float rv[256];
    __shared__ int   ri[256];
    int t = threadIdx.x;
    float px = pos[(b * Lc_ + i) * 3 + 0];
    float py = pos[(b * Lc_ + i) * 3 + 1];
    float pz = pos[(b * Lc_ + i) * 3 + 2];
    for (int j = t; j < Lc_; j += 256) {
        float dx = px - pos[(b * Lc_ + j) * 3 + 0];
        float dy = py - pos[(b * Lc_ + j) * 3 + 1];
        float dz = pz - pos[(b * Lc_ + j) * 3 + 2];
        float dist = sqrtf(dx*dx + dy*dy + dz*dz);
        cl[j] = -dist + 3.0f * noise[((long)b * Lc_ + i) * Lc_ + j];
    }
    __syncthreads();
    for (int k = 0; k < NBRc; ++k) {
        float bv = -3.0e38f; int bi = Lc_;
        for (int j = t; j < Lc_; j += 256) {
            float v = cl[j];
            if (v > bv || (v == bv && j < bi)) { bv = v; bi = j; }
        }
        rv[t] = bv; ri[t] = bi;
        __syncthreads();
        for (int s = 128; s > 0; s >>= 1) {
            if (t < s) {
                if (rv[t + s] > rv[t] || (rv[t + s] == rv[t] && ri[t + s] < ri[t])) {
                    rv[t] = rv[t + s]; ri[t] = ri[t + s];
                }
            }
            __syncthreads();
        }
        if (t == 0) {
            nbr[(b * Lc_ + i) * NBRc + k] = b * Lc_ + ri[0];
            cl[ri[0]] = -3.0e38f;
        }
        __syncthreads();
    }
}

// ------------------------------------------------------------ edge feats ----
__global__ void edge_kernel(const float* __restrict__ pos,
                            const float* __restrict__ R,
                            const int* __restrict__ nbr,
                            float* __restrict__ edge) {
    int idx = blockIdx.x * blockDim.x + threadIdx.x;
    if (idx >= Nn * NBRc) return;
    int n = idx / NBRc;
    int m = nbr[idx];
    float dv[3];
    for (int c = 0; c < 3; ++c) dv[c] = pos[m * 3 + c] - pos[n * 3 + c];
    float d = sqrtf(dv[0]*dv[0] + dv[1]*dv[1] + dv[2]*dv[2]);
    float* e = edge + (long)idx * 32;
    const float sigma = MAXDc / (float)KERc;            // 1.25
    const float mustep = MAXDc / (float)(KERc - 1);     // linspace(0,20,16)
    for (int tkk = 0; tkk < KERc; ++tkk) {
        float z = (d - mustep * tkk) / sigma;
        e[tkk] = expf(-z * z);
    }
    float invd = 1.0f / (d + 1e-8f);
    float u[3] = { dv[0] * invd, dv[1] * invd, dv[2] * invd };
    float Rn[9], Rm[9];
    for (int q = 0; q < 9; ++q) { Rn[q] = R[n * 9 + q]; Rm[q] = R[m * 9 + q]; }
    for (int c = 0; c < 3; ++c)
        e[16 + c] = Rn[0*3 + c]*u[0] + Rn[1*3 + c]*u[1] + Rn[2*3 + c]*u[2];
    for (int c = 0; c < 3; ++c)
        for (int dd = 0; dd < 3; ++dd)
            e[19 + c*3 + dd] = Rn[0*3+c]*Rm[0*3+dd] + Rn[1*3+c]*Rm[1*3+dd] + Rn[2*3+c]*Rm[2*3+dd];
    e[28] = (float)(m - n);
    e[29] = 0.f; e[30] = 0.f; e[31] = 0.f;   // pad to K=160 for WMMA
}

// ------------------------------------------------------------- h = init -----
__global__ void h_init_kernel(const float* __restrict__ ew,
                              const float* __restrict__ eb,
                              float* __restrict__ h) {
    int idx = blockIdx.x * blockDim.x + threadIdx.x;
    if (idx >= Nn * SZ) return;
    int c = idx & (SZ - 1);
    float s = eb[c];
    for (int r = 0; r < 27; ++r) s += ew[r * SZ + c];
    h[idx] = s;
}

// ---------------------------------------------------- pad wk/wv to K=160 ----
__global__ void padw_kernel(const float* __restrict__ wk, const float* __restrict__ wv,
                            float* __restrict__ wkp, float* __restrict__ wvp) {
    int idx = blockIdx.x * blockDim.x + threadIdx.x;
    if (idx >= KVKP * SZ) return;
    int r = idx >> 7;
    wkp[idx] = (r < KVK) ? wk[idx] : 0.f;
    wvp[idx] = (r < KVK) ? wv[idx] : 0.f;
}

// ----------------------------------------- generic 128-K WMMA GEMM epilog ---
// MODE 0: C = A@W            (Q projection)
// MODE 1: C = relu(A@W + b)  (MLP hidden)
// MODE 2: C = A@W + b + res  (MLP out, residual)
// MODE 3: C = A@W + res      (O projection, residual)
template <int MODE>
__global__ void gemm128_kernel(const float* __restrict__ A, const float* __restrict__ W,
                               const float* __restrict__ bias, const float* __restrict__ resid,
                               float* __restrict__ C) {
    __shared__ float As[16][129];
    int t = threadIdx.x;
    int row0 = blockIdx.x * 16;
    for (int idx = t; idx < 16 * 128; idx += 256)
        As[idx >> 7][idx & 127] = A[(row0 + (idx >> 7)) * SZ + (idx & 127)];
    __syncthreads();
    int w = t >> 5, lane = t & 31;
    int half = lane >> 4, ln = lane & 15;
    int col0 = w * 16;
    f32x8 acc = {};
    for (int k = 0; k < 128; k += 4) {
        int kk = k + 2 * half;
        f32x2 a; a.x = As[ln][kk]; a.y = As[ln][kk + 1];
        f32x2 bb; bb.x = W[kk * SZ + col0 + ln]; bb.y = W[(kk + 1) * SZ + col0 + ln];
        acc = wmma4(a, bb, acc);
    }
    for (int v = 0; v < 8; ++v) {
        int m = v + 8 * half;
        int row = row0 + m, col = col0 + ln;
        float val = acc[v];
        if (MODE == 1 || MODE == 2) val += bias[col];
        if (MODE == 1) val = fmaxf(val, 0.f);
        if (MODE == 2 || MODE == 3) val += resid[row * SZ + col];
        C[row * SZ + col] = val;
    }
}

// ----------------------- fused K/V projection (WMMA) + softmax attention ----
__global__ void attn_kernel(const float* __restrict__ h, const float* __restrict__ qb,
                            const int* __restrict__ nbr, const float* __restrict__ edge,
                            const float* __restrict__ wkp, const float* __restrict__ wvp,
                            float* __restrict__ ob) {
    int n = blockIdx.x;
    __shared__ float As[16][161];
    __shared__ float Ks[16][129];
    __shared__ float Vs[16][129];
    __shared__ float qs[128];
    __shared__ float att[HEADSc][NBRc];
    int t = threadIdx.x;
    // stage A = [h[nbr] | edge | 0pad], row 15 = zeros
    for (int idx = t; idx < 16 * KVKP; idx += 256) {
        int r = idx / KVKP, cc = idx % KVKP;
        float v = 0.f;
        if (r < NBRc) {
            if (cc < 128)       v = h[nbr[n * NBRc + r] * SZ + cc];
            else if (cc < 157)  v = edge[((long)(n * NBRc + r)) * 32 + (cc - 128)];
        }
        As[r][cc] = v;
    }
    if (t < 128) qs[t] = qb[n * SZ + t];
    __syncthreads();
    int w = t >> 5, lane = t & 31;
    int half = lane >> 4, ln = lane & 15;
    int col0 = w * 16;
    f32x8 ck = {}, cv = {};
    for (int k = 0; k < KVKP; k += 4) {
        int kk = k + 2 * half;
        f32x2 a;  a.x  = As[ln][kk];            a.y  = As[ln][kk + 1];
        f32x2 bk; bk.x = wkp[kk * SZ + col0 + ln]; bk.y = wkp[(kk + 1) * SZ + col0 + ln];
        f32x2 bv; bv.x = wvp[kk * SZ + col0 + ln]; bv.y = wvp[(kk + 1) * SZ + col0 + ln];
        ck = wmma4(a, bk, ck);
        cv = wmma4(a, bv, cv);
    }
    for (int v = 0; v < 8; ++v) {
        int m = v + 8 * half;
        Ks[m][col0 + ln] = ck[v];
        Vs[m][col0 + ln] = cv[v];
    }
    __syncthreads();
    if (t < HEADSc * NBRc) {
        int hh = t / NBRc, j = t % NBRc;
        float s = 0.f;
        for (int d = 0; d < AHc; ++d) s += qs[hh * AHc + d] * Ks[j][hh * AHc + d];
        att[hh][j] = s * 0.25f;   // 1/sqrt(16)
    }
    __syncthreads();
    if (t < HEADSc) {
        float mx = -3.0e38f;
        for (int j = 0; j < NBRc; ++j) mx = fmaxf(mx, att[t][j]);
        float sm = 0.f;
        for (int j = 0; j < NBRc; ++j) { float e = expf(att[t][j] - mx); att[t][j] = e; sm += e; }
        float inv = 1.f / sm;
        for (int j = 0; j < NBRc; ++j) att[t][j] *= inv;
    }
    __syncthreads();
    if (t < 128) {
        int hh = t >> 4;
        float s = 0.f;
        for (int j = 0; j < NBRc; ++j) s += att[hh][j] * Vs[j][t];
        ob[n * SZ + t] = s;
    }
}

// ----------------------------------------------------------- NCH layout -----
__global__ void to_nchw_kernel(const float* __restrict__ h, float* __restrict__ x) {
    int idx = blockIdx.x * blockDim.x + threadIdx.x;
    if (idx >= Nn * SZ) return;
    int c = idx & 127, nl = idx >> 7;
    int b = nl >> 9, tt = nl & 511;
    x[(b * SZ + c) * Lc_ + tt] = h[idx];
}

// ------------------------------------------- pool weight -> (384,128) -------
__global__ void poolprep_kernel(const float* __restrict__ pw, float* __restrict__ wpt) {
    int idx = blockIdx.x * blockDim.x + threadIdx.x;
    if (idx >= 384 * 128) return;
    int k = idx >> 7, o = idx & 127;
    int s = k / 128, ic = k % 128;
    wpt[idx] = pw[(o * 128 + ic) * 3 + s];
}

// ------------------------- conv1d(k=3,pad=1) as WMMA GEMM + leaky + res -----
__global__ void conv_kernel(const float* __restrict__ x, const float* __restrict__ wpt,
                            const float* __restrict__ pb, float* __restrict__ y, int Lc) {
    int tilesPerB = Lc >> 4;
    int b = blockIdx.x / tilesPerB;
    int t0 = (blockIdx.x % tilesPerB) * 16;
    __shared__ float As[16][385];
    int t = threadIdx.x;
    for (int idx = t; idx < 16 * 384; idx += 256) {
        int r = idx / 384, k = idx % 384;
        int s = k >> 7, ic = k & 127;
        int tt = t0 + r - 1 + s;
        float v = 0.f;
        if (tt >= 0 && tt < Lc) v = x[(b * SZ + ic) * Lc + tt];
        As[r][k] = v;
    }
    __syncthreads();
    int w = t >> 5, lane = t & 31;
    int half = lane >> 4, ln = lane & 15;
    int col0 = w * 16;
    f32x8 acc = {};
    for (int k = 0; k < 384; k += 4) {
        int kk = k + 2 * half;
        f32x2 a;  a.x  = As[ln][kk];              a.y  = As[ln][kk + 1];
        f32x2 bb; bb.x = wpt[kk * SZ + col0 + ln]; bb.y = wpt[(kk + 1) * SZ + col0 + ln];
        acc = wmma4(a, bb, acc);
    }
    for (int v = 0; v < 8; ++v) {
        int m = v + 8 * half;
        int row = t0 + m, o = col0 + ln;
        float yv = acc[v] + pb[o];
        float act = yv > 0.f ? yv : 0.01f * yv;
        y[(b * SZ + o) * Lc + row] = x[(b * SZ + o) * Lc + row] + act;
    }
}

__global__ void maxpool_kernel(const float* __restrict__ y, float* __restrict__ x, int Lh) {
    int idx = blockIdx.x * blockDim.x + threadIdx.x;
    if (idx >= Bc * SZ * Lh) return;
    int tt = idx % Lh, bc = idx / Lh;
    x[bc * Lh + tt] = fmaxf(y[bc * (2 * Lh) + 2 * tt], y[bc * (2 * Lh) + 2 * tt + 1]);
}

// --------------------------------------------------------------- energy -----
__global__ void energy_kernel(const float* __restrict__ x, const float* __restrict__ ew,
                              const float* __restrict__ eb, float* __restrict__ out) {
    int b = blockIdx.x;
    __shared__ float red[128];
    int t = threadIdx.x;
    float s = 0.f;
    for (int tt = 0; tt < 32; ++tt) s += x[(b * SZ + t) * 32 + tt];
    red[t] = s * ew[t];
    __syncthreads();
    for (int st = 64; st > 0; st >>= 1) {
        if (t < st) red[t] += red[t + st];
        __syncthreads();
    }
    if (t == 0) out[b] = red[0] + eb[0];
}

// ---------------------------------------------------------------------------
extern "C" void kernel_launch(void* const* d_in, const int* in_sizes, int n_in,
                              void* d_out, int out_size, void* d_ws, size_t ws_size,
                              hipStream_t stream) {
    (void)in_sizes; (void)n_in; (void)out_size; (void)ws_size;
    const float* tert  = (const float*)d_in[0];
    const float* noise = (const float*)d_in[3];
    const float* ew    = (const float*)d_in[4];
    const float* ebv   = (const float*)d_in[5];
    const float* wq    = (const float*)d_in[6];
    const float* wk    = (const float*)d_in[7];
    const float* wv    = (const float*)d_in[8];
    const float* wo    = (const float*)d_in[9];
    const float* w1    = (const float*)d_in[10];
    const float* b1    = (const float*)d_in[11];
    const float* w2    = (const float*)d_in[12];
    const float* b2    = (const float*)d_in[13];
    const float* w3    = (const float*)d_in[14];
    const float* b3    = (const float*)d_in[15];
    const float* pw    = (const float*)d_in[16];
    const float* pb    = (const float*)d_in[17];
    const float* enw   = (const float*)d_in[18];
    const float* enb   = (const float*)d_in[19];
    float* out = (float*)d_out;

    char* ws = (char*)d_ws;
    size_t off = 0;
    auto alloc = [&](size_t bytes) -> void* {
        void* p = ws + off;
        off += (bytes + 255) & ~(size_t)255;
        return p;
    };
    float* pos  = (float*)alloc((size_t)Nn * 3 * 4);
    float* R    = (float*)alloc((size_t)Nn * 9 * 4);
    int*   nbr  = (int*)  alloc((size_t)Nn * NBRc * 4);
    float* edge = (float*)alloc((size_t)Nn * NBRc * 32 * 4);
    float* h    = (float*)alloc((size_t)Nn * SZ * 4);
    float* qb   = (float*)alloc((size_t)Nn * SZ * 4);
    float* ob   = (float*)alloc((size_t)Nn * SZ * 4);
    float* t1   = (float*)alloc((size_t)Nn * SZ * 4);   // also pooling bufA
    float* t2   = (float*)alloc((size_t)Nn * SZ * 4);   // also pooling bufB
    float* wkp  = (float*)alloc((size_t)KVKP * SZ * 4);
    float* wvp  = (float*)alloc((size_t)KVKP * SZ * 4);
    float* wpt  = (float*)alloc((size_t)384 * 128 * 4);
    float* bufA = t1;
    float* bufB = t2;

    frames_kernel<<<(Nn + 255) / 256, 256, 0, stream>>>(tert, pos, R);
    topk_kernel<<<dim3(Lc_, Bc), 256, 0, stream>>>(pos, noise, nbr);
    edge_kernel<<<(Nn * NBRc + 255) / 256, 256, 0, stream>>>(pos, R, nbr, edge);
    h_init_kernel<<<(Nn * SZ + 255) / 256, 256, 0, stream>>>(ew, ebv, h);

    for (int l = 0; l < DEPTHc; ++l) {
        padw_kernel<<<(KVKP * SZ + 255) / 256, 256, 0, stream>>>(
            wk + (size_t)l * KVK * SZ, wv + (size_t)l * KVK * SZ, wkp, wvp);
        gemm128_kernel<0><<<Nn / 16, 256, 0, stream>>>(
            h, wq + (size_t)l * SZ * SZ, nullptr, nullptr, qb);
        attn_kernel<<<Nn, 256, 0, stream>>>(h, qb, nbr, edge, wkp, wvp, ob);
        gemm128_kernel<3><<<Nn / 16, 256, 0, stream>>>(
            ob, wo + (size_t)l * SZ * SZ, nullptr, h, h);
        gemm128_kernel<1><<<Nn / 16, 256, 0, stream>>>(
            h, w1 + (size_t)l * SZ * SZ, b1 + (size_t)l * SZ, nullptr, t1);
        gemm128_kernel<1><<<Nn / 16, 256, 0, stream>>>(
            t1, w2 + (size_t)l * SZ * SZ, b2 + (size_t)l * SZ, nullptr, t2);
        gemm128_kernel<2><<<Nn / 16, 256, 0, stream>>>(
            t2, w3 + (size_t)l * SZ * SZ, b3 + (size_t)l * SZ, h, h);
    }

    to_nchw_kernel<<<(Nn * SZ + 255) / 256, 256, 0, stream>>>(h, bufA);
    int Lcur = Lc_;
    for (int s = 0; s < 4; ++s) {
        poolprep_kernel<<<(384 * 128 + 255) / 256, 256, 0, stream>>>(
            pw + (size_t)s * 128 * 128 * 3, wpt);
        conv_kernel<<<Bc * (Lcur / 16), 256, 0, stream>>>(bufA, wpt, pb + (size_t)s * SZ, bufB, Lcur);
        maxpool_kernel<<<(Bc * SZ * (Lcur / 2) + 255) / 256, 256, 0, stream>>>(bufB, bufA, Lcur / 2);
        Lcur /= 2;
    }
    energy_kernel<<<Bc, 128, 0, stream>>>(bufA, enw, enb, out);
}